// NetworkAwareQuantumExpert_34136400069004
// MI455X (gfx1250) — compile-verified
//
#include <hip/hip_runtime.h>
#include <math.h>

#define DEVINL __device__ __forceinline__

typedef __attribute__((ext_vector_type(2))) float v2f;
typedef __attribute__((ext_vector_type(8))) float v8f;

constexpr int NB = 16;
constexpr int NT = 363;
constexpr int NC = 180;
constexpr int BT = NB * NT;     // 5808 rows, = 363 * 16 (exact 16-row tiles)
constexpr int GPB = 6;          // quantum: blocks per batch element
constexpr int NWV = GPB * 8;    // 48 waves per batch element

// ---------------------------------------------------------------- utilities
DEVINL float gelu_f(float x) { return 0.5f * x * (1.0f + erff(x * 0.7071067811865475f)); }

DEVINL float wred32(float v) {
  v += __shfl_xor(v, 16, 32);
  v += __shfl_xor(v, 8, 32);
  v += __shfl_xor(v, 4, 32);
  v += __shfl_xor(v, 2, 32);
  v += __shfl_xor(v, 1, 32);
  return v;
}

// ================================================================ quantum sim
// 8 qubits, 256 complex amps per state, one wave32 per state.
// amplitude index idx[7:0]:  bits 2:0 -> register r (8 complex per lane),
//                            bits 7:3 -> lane id.   wire w <-> bit (7-w).

struct CS { float c0, s0, c1, s1; };

template<int P> DEVINL void getcs(const CS& cs, float& c, float& s) {
  if constexpr (P < 32) { c = __shfl(cs.c0, P, 32); s = __shfl(cs.s0, P, 32); }
  else                  { c = __shfl(cs.c1, P - 32, 32); s = __shfl(cs.s1, P - 32, 32); }
}

template<int TB> DEVINL void apply_ry(float re[8], float im[8], float c, float s, int lane) {
  if constexpr (TB < 3) {
    constexpr int m = 1 << TB;
#pragma unroll
    for (int r = 0; r < 8; ++r) {
      if ((r & m) == 0) {
        int r1 = r | m;
        float a0r = re[r], a1r = re[r1], a0i = im[r], a1i = im[r1];
        re[r]  = c * a0r - s * a1r;   re[r1] = s * a0r + c * a1r;
        im[r]  = c * a0i - s * a1i;   im[r1] = s * a0i + c * a1i;
      }
    }
  } else {
    constexpr int lm = 1 << (TB - 3);
    float sg = (lane & lm) ? s : -s;
#pragma unroll
    for (int r = 0; r < 8; ++r) {
      float pr = __shfl_xor(re[r], lm, 32);
      float pi = __shfl_xor(im[r], lm, 32);
      re[r] = fmaf(sg, pr, c * re[r]);
      im[r] = fmaf(sg, pi, c * im[r]);
    }
  }
}

// CRX: control bit CB, target bit TB.  Rx = [[cs, -i*sn],[-i*sn, cs]]
// symmetric per-element form: n_re = cs*a_re + sn*partner_im ; n_im = cs*a_im - sn*partner_re
template<int CB, int TB> DEVINL void apply_crx(float re[8], float im[8], float cc, float sn, int lane) {
  if constexpr (TB < 3) {
    constexpr int tm = 1 << TB;
    bool lcond = true;
    if constexpr (CB >= 3) lcond = ((lane >> (CB - 3)) & 1) != 0;
#pragma unroll
    for (int r = 0; r < 8; ++r) {
      if ((r & tm) == 0) {
        int r1 = r | tm;
        bool take = true;
        if constexpr (CB < 3) take = ((r >> CB) & 1) != 0;
        if (take) {
          float n0r = cc * re[r]  + sn * im[r1];
          float n0i = cc * im[r]  - sn * re[r1];
          float n1r = cc * re[r1] + sn * im[r];
          float n1i = cc * im[r1] - sn * re[r];
          if constexpr (CB < 3) { re[r] = n0r; im[r] = n0i; re[r1] = n1r; im[r1] = n1i; }
          else {
            re[r]  = lcond ? n0r : re[r];   im[r]  = lcond ? n0i : im[r];
            re[r1] = lcond ? n1r : re[r1];  im[r1] = lcond ? n1i : im[r1];
          }
        }
      }
    }
  } else {
    constexpr int lm = 1 << (TB - 3);
    bool lcond = true;
    if constexpr (CB >= 3) lcond = ((lane >> (CB - 3)) & 1) != 0;
#pragma unroll
    for (int r = 0; r < 8; ++r) {
      bool rcond = true;
      if constexpr (CB < 3) rcond = ((r >> CB) & 1) != 0;
      if (rcond) {   // compile-time uniform, shuffles stay wave-uniform
        float pr = __shfl_xor(re[r], lm, 32);
        float pi = __shfl_xor(im[r], lm, 32);
        float nr = cc * re[r] + sn * pi;
        float ni = cc * im[r] - sn * pr;
        if constexpr (CB >= 3) { re[r] = lcond ? nr : re[r]; im[r] = lcond ? ni : im[r]; }
        else                   { re[r] = nr; im[r] = ni; }
      }
    }
  }
}

#define FOR8(F) F(0) F(1) F(2) F(3) F(4) F(5) F(6) F(7)

// One sim14 layer (32 params starting at compile-time base PB):
// RY(q) p+q | CRX(q -> q+1) p+8+q | RY(q) p+16+q | CRX(q -> q-1) p+24+q
template<int PB> DEVINL void sim_layer(float re[8], float im[8], const CS& cs, int lane) {
  float c, s;
#define G_RY1(q)  { getcs<PB + (q)>(cs, c, s);      apply_ry<7 - (q)>(re, im, c, s, lane); }
#define G_CRX1(q) { getcs<PB + 8 + (q)>(cs, c, s);  apply_crx<7 - (q), 7 - (((q) + 1) & 7)>(re, im, c, s, lane); }
#define G_RY2(q)  { getcs<PB + 16 + (q)>(cs, c, s); apply_ry<7 - (q)>(re, im, c, s, lane); }
#define G_CRX2(q) { getcs<PB + 24 + (q)>(cs, c, s); apply_crx<7 - (q), 7 - (((q) + 7) & 7)>(re, im, c, s, lane); }
  FOR8(G_RY1)
  FOR8(G_CRX1)
  FOR8(G_RY2)
  FOR8(G_CRX2)
#undef G_RY1
#undef G_CRX1
#undef G_RY2
#undef G_CRX2
}

template<int TB> DEVINL void pauli(const float re[8], const float im[8], int lane,
                                   float& XR, float& XI, float& ZZ) {
  float xr = 0.f, xi = 0.f, zz = 0.f;
  if constexpr (TB < 3) {
    constexpr int m = 1 << TB;
#pragma unroll
    for (int r = 0; r < 8; ++r) {
      if ((r & m) == 0) {
        int r1 = r | m;
        xr += re[r] * re[r1] + im[r] * im[r1];
        xi += re[r] * im[r1] - im[r] * re[r1];
        zz += re[r] * re[r] + im[r] * im[r] - re[r1] * re[r1] - im[r1] * im[r1];
      }
    }
  } else {
    constexpr int lm = 1 << (TB - 3);
    bool hi = (lane & lm) != 0;
#pragma unroll
    for (int r = 0; r < 8; ++r) {
      float pr = __shfl_xor(re[r], lm, 32);
      float pi = __shfl_xor(im[r], lm, 32);
      float cr = re[r] * pr + im[r] * pi;
      float ci = re[r] * pi - im[r] * pr;
      float mg = re[r] * re[r] + im[r] * im[r];
      xr += hi ? 0.f : cr;
      xi += hi ? 0.f : ci;
      zz += hi ? -mg : mg;
    }
  }
  XR = wred32(xr); XI = wred32(xi); ZZ = wred32(zz);
}

// ================================================================ K1: stage1 MLP per 15-channel group
__global__ void __launch_bounds__(256) k_stage1(const float* __restrict__ x,
                                                const float* __restrict__ net_w,
                                                const float* __restrict__ net_b,
                                                float* __restrict__ h) {
  int gid = blockIdx.x * 256 + threadIdx.x;
  if (gid >= BT * 96) return;
  int row = gid / 96, col = gid % 96;
  int k = col >> 3, d = col & 7;
  float acc = net_b[k * 8 + d];
  const float* xr = x + (size_t)row * NC + k * 15;
  const float* wr = net_w + (size_t)k * 15 * 8 + d;
#pragma unroll
  for (int i = 0; i < 15; ++i) acc = fmaf(xr[i], wr[i * 8], acc);
  h[gid] = gelu_f(acc);
}

// ================================================================ K2: depthwise conv (dilated, k=7, zero pad)
__global__ void __launch_bounds__(256) k_dw(const float* __restrict__ h,
                                            const float* __restrict__ dw_w,
                                            const float* __restrict__ dw_b,
                                            int dil, float* __restrict__ out) {
  int gid = blockIdx.x * 256 + threadIdx.x;
  if (gid >= BT * 96) return;
  int row = gid / 96, c = gid % 96;
  int b = row / NT, t = row % NT;
  const float* wc = dw_w + c * 7;
  float acc = dw_b[c];
#pragma unroll
  for (int j = 0; j < 7; ++j) {
    int tt = t + (j - 3) * dil;
    if (tt >= 0 && tt < NT) acc = fmaf(h[((size_t)b * NT + tt) * 96 + c], wc[j], acc);
  }
  out[gid] = acc;
}

// ================================================================ K: transpose pw weights (o,c) -> (c,o)
__global__ void __launch_bounds__(256) k_tr96(const float* __restrict__ w, float* __restrict__ wt) {
  int gid = blockIdx.x * 256 + threadIdx.x;
  if (gid >= 96 * 96) return;
  int c = gid / 96, o = gid % 96;
  wt[c * 96 + o] = w[o * 96 + c];
}

// ================================================================ K3: pointwise 96x96 GEMM via f32 WMMA
__global__ void __launch_bounds__(192) k_pw(const float* __restrict__ A,    // (BT,96)
                                            const float* __restrict__ Bkn,  // (96,96) K-major
                                            const float* __restrict__ bias,
                                            float* __restrict__ Out) {
  __shared__ float ldsA[16 * 97];
  int m0 = blockIdx.x * 16;
  int tid = threadIdx.x;
  for (int i = tid; i < 16 * 96; i += 192) ldsA[(i / 96) * 97 + (i % 96)] = A[(size_t)(m0 + i / 96) * 96 + (i % 96)];
  __syncthreads();
  int lane = tid & 31, wave = tid >> 5;
  int n0 = wave * 16, ml = lane & 15, koff = (lane >> 4) * 2;
  v8f acc = {0.f, 0.f, 0.f, 0.f, 0.f, 0.f, 0.f, 0.f};
  for (int k = 0; k < 96; k += 4) {
    v2f a, b;
    a.x = ldsA[ml * 97 + k + koff];
    a.y = ldsA[ml * 97 + k + koff + 1];
    const float* bp = Bkn + (size_t)(k + koff) * 96 + n0 + ml;
    b.x = bp[0];
    b.y = bp[96];
    acc = __builtin_amdgcn_wmma_f32_16x16x4_f32(false, a, false, b, (short)0, acc, false, false);
  }
  float bb = bias[n0 + ml];
  int mhi = (lane >> 4) * 8;
#pragma unroll
  for (int r = 0; r < 8; ++r)
    Out[(size_t)(m0 + r + mhi) * 96 + n0 + ml] = acc[r] + bb;
}

// ================================================================ K4: group-norm stats (B x 8 groups over 12ch x T)
__global__ void __launch_bounds__(256) k_gnstats(const float* __restrict__ y, float* __restrict__ gn) {
  int lane = threadIdx.x & 31;
  int wid = blockIdx.x * 8 + (threadIdx.x >> 5);   // 0..127
  int b = wid >> 3, g = wid & 7;
  float s1 = 0.f, s2 = 0.f;
  for (int i = lane; i < NT * 12; i += 32) {
    int t = i / 12, c = g * 12 + (i % 12);
    float v = y[((size_t)b * NT + t) * 96 + c];
    s1 += v; s2 += v * v;
  }
  s1 = wred32(s1); s2 = wred32(s2);
  if (lane == 0) {
    float mu = s1 / (float)(NT * 12);
    float var = s2 / (float)(NT * 12) - mu * mu;
    gn[wid * 2 + 0] = mu;
    gn[wid * 2 + 1] = rsqrtf(var + 1e-5f);
  }
}

// ================================================================ K5: apply GN + affine + gelu, write into cat slice
__global__ void __launch_bounds__(256) k_gnapply(const float* __restrict__ y,
                                                 const float* __restrict__ gn,
                                                 const float* __restrict__ gg,
                                                 const float* __restrict__ gb,
                                                 int brOff, float* __restrict__ cat) {
  int gid = blockIdx.x * 256 + threadIdx.x;
  if (gid >= BT * 96) return;
  int row = gid / 96, c = gid % 96;
  int b = row / NT, g = c / 12;
  float mu = gn[(b * 8 + g) * 2 + 0];
  float rs = gn[(b * 8 + g) * 2 + 1];
  float v = (y[gid] - mu) * rs;
  cat[(size_t)row * 288 + brOff + c] = gelu_f(v * gg[c] + gb[c]);
}

// ================================================================ K6: merge GEMM (K=288,N=128) + bias + LN + gelu
__global__ void __launch_bounds__(256) k_merge(const float* __restrict__ A,    // (BT,288)
                                               const float* __restrict__ Bkn,  // (288,128)
                                               const float* __restrict__ bias,
                                               const float* __restrict__ lng,
                                               const float* __restrict__ lnb,
                                               float* __restrict__ Out) {
  __shared__ float ldsA[16 * 289];
  __shared__ float zt[16 * 130];
  int m0 = blockIdx.x * 16;
  int tid = threadIdx.x;
  for (int i = tid; i < 16 * 288; i += 256) ldsA[(i / 288) * 289 + (i % 288)] = A[(size_t)(m0 + i / 288) * 288 + (i % 288)];
  __syncthreads();
  int lane = tid & 31, wave = tid >> 5;
  int n0 = wave * 16, ml = lane & 15, koff = (lane >> 4) * 2;
  v8f acc = {0.f, 0.f, 0.f, 0.f, 0.f, 0.f, 0.f, 0.f};
  for (int k = 0; k < 288; k += 4) {
    v2f a, b;
    a.x = ldsA[ml * 289 + k + koff];
    a.y = ldsA[ml * 289 + k + koff + 1];
    const float* bp = Bkn + (size_t)(k + koff) * 128 + n0 + ml;
    b.x = bp[0];
    b.y = bp[128];
    acc = __builtin_amdgcn_wmma_f32_16x16x4_f32(false, a, false, b, (short)0, acc, false, false);
  }
  float bb = bias[n0 + ml];
  int mhi = (lane >> 4) * 8;
#pragma unroll
  for (int r = 0; r < 8; ++r) zt[(r + mhi) * 130 + n0 + ml] = acc[r] + bb;
  __syncthreads();
  // LayerNorm over 128 per row: 16 threads per row, 8 cols each
  int row = tid >> 4, j = tid & 15;
  float s1 = 0.f, s2 = 0.f;
#pragma unroll
  for (int u = 0; u < 8; ++u) { float v = zt[row * 130 + j * 8 + u]; s1 += v; s2 += v * v; }
  s1 += __shfl_xor(s1, 1, 32); s2 += __shfl_xor(s2, 1, 32);
  s1 += __shfl_xor(s1, 2, 32); s2 += __shfl_xor(s2, 2, 32);
  s1 += __shfl_xor(s1, 4, 32); s2 += __shfl_xor(s2, 4, 32);
  s1 += __shfl_xor(s1, 8, 32); s2 += __shfl_xor(s2, 8, 32);
  float mu = s1 * (1.f / 128.f);
  float var = s2 * (1.f / 128.f) - mu * mu;
  float rs = rsqrtf(var + 1e-5f);
#pragma unroll
  for (int u = 0; u < 8; ++u) {
    int c = j * 8 + u;
    float v = (zt[row * 130 + c] - mu) * rs * lng[c] + lnb[c];
    Out[(size_t)(m0 + row) * 128 + c] = gelu_f(v);
  }
}

// ================================================================ K7: qproj GEMM (K=128,N=64) + sigmoid*2pi
__global__ void __launch_bounds__(128) k_qproj(const float* __restrict__ A,    // (BT,128)
                                               const float* __restrict__ Bkn,  // (128,64)
                                               const float* __restrict__ bias,
                                               float* __restrict__ Out) {
  __shared__ float ldsA[16 * 129];
  int m0 = blockIdx.x * 16;
  int tid = threadIdx.x;
  for (int i = tid; i < 16 * 128; i += 128) ldsA[(i / 128) * 129 + (i % 128)] = A[(size_t)(m0 + i / 128) * 128 + (i % 128)];
  __syncthreads();
  int lane = tid & 31, wave = tid >> 5;
  int n0 = wave * 16, ml = lane & 15, koff = (lane >> 4) * 2;
  v8f acc = {0.f, 0.f, 0.f, 0.f, 0.f, 0.f, 0.f, 0.f};
  for (int k = 0; k < 128; k += 4) {
    v2f a, b;
    a.x = ldsA[ml * 129 + k + koff];
    a.y = ldsA[ml * 129 + k + koff + 1];
    const float* bp = Bkn + (size_t)(k + koff) * 64 + n0 + ml;
    b.x = bp[0];
    b.y = bp[64];
    acc = __builtin_amdgcn_wmma_f32_16x16x4_f32(false, a, false, b, (short)0, acc, false, false);
  }
  float bb = bias[n0 + ml];
  int mhi = (lane >> 4) * 8;
#pragma unroll
  for (int r = 0; r < 8; ++r) {
    float z = acc[r] + bb;
    Out[(size_t)(m0 + r + mhi) * 64 + n0 + ml] = 6.283185307179586f / (1.f + expf(-z));
  }
}

// ================================================================ K8: quantum polynomial feature map (wave per state)
__global__ void __launch_bounds__(256) k_quantum(const float* __restrict__ tp,     // (BT,64)
                                                 const float* __restrict__ pc4,    // (4,)
                                                 const float* __restrict__ mixre,
                                                 const float* __restrict__ mixim,
                                                 float* __restrict__ part) {       // (B,NWV,512)
  int lane = threadIdx.x & 31;
  int wave = threadIdx.x >> 5;
  int b = blockIdx.x / GPB;
  int wid = (blockIdx.x % GPB) * 8 + wave;     // 0..NWV-1
  float pc0 = pc4[0], pc1 = pc4[1], pc2 = pc4[2], pc3 = pc4[3];
  float amR[8], amI[8];
#pragma unroll
  for (int r = 0; r < 8; ++r) { amR[r] = 0.f; amI[r] = 0.f; }
  for (int t = wid; t < NT; t += NWV) {
    size_t bt = (size_t)b * NT + t;
    const float* tpp = tp + bt * 64;
    CS cs;
    __sincosf(0.5f * tpp[lane],      &cs.s0, &cs.c0);
    __sincosf(0.5f * tpp[32 + lane], &cs.s1, &cs.c1);
    float re[8], im[8], aR[8], aI[8];
#pragma unroll
    for (int r = 0; r < 8; ++r) { re[r] = 0.f; im[r] = 0.f; aR[r] = 0.f; aI[r] = 0.f; }
    re[0] = (lane == 0) ? 1.0f : 0.0f;   // |0...0>
    aR[0] = (lane == 0) ? pc0  : 0.0f;   // pc[0] * base
#pragma unroll
    for (int d = 1; d <= 3; ++d) {
      sim_layer<0>(re, im, cs, lane);
      sim_layer<32>(re, im, cs, lane);
      float pcd = (d == 1) ? pc1 : ((d == 2) ? pc2 : pc3);
#pragma unroll
      for (int r = 0; r < 8; ++r) { aR[r] = fmaf(pcd, re[r], aR[r]); aI[r] = fmaf(pcd, im[r], aI[r]); }
    }
    float mr = mixre[t], mi = mixim[t];
#pragma unroll
    for (int r = 0; r < 8; ++r) {
      amR[r] += mr * aR[r] - mi * aI[r];
      amI[r] += mr * aI[r] + mi * aR[r];
    }
  }
  float* pb = part + ((size_t)b * NWV + wid) * 512 + lane * 16;
#pragma unroll
  for (int r = 0; r < 8; ++r) { pb[2 * r] = amR[r]; pb[2 * r + 1] = amI[r]; }
}

// ================================================================ K9: reduce partials, normalize, qff layer, Pauli, temporal proj
__global__ void __launch_bounds__(32) k_qfinal(const float* __restrict__ part,
                                               const float* __restrict__ qff,
                                               const float* __restrict__ qout_w,  // (24,128)
                                               const float* __restrict__ qout_b,
                                               float* __restrict__ temporal) {    // (B,128)
  int b = blockIdx.x;
  int lane = threadIdx.x;
  float re[8], im[8];
#pragma unroll
  for (int r = 0; r < 8; ++r) { re[r] = 0.f; im[r] = 0.f; }
  const float* pb = part + (size_t)b * NWV * 512 + lane * 16;
  for (int w = 0; w < NWV; ++w) {
    const float* p = pb + (size_t)w * 512;
#pragma unroll
    for (int r = 0; r < 8; ++r) { re[r] += p[2 * r]; im[r] += p[2 * r + 1]; }
  }
  float ss = 0.f;
#pragma unroll
  for (int r = 0; r < 8; ++r) ss += re[r] * re[r] + im[r] * im[r];
  ss = wred32(ss);
  float inv = 1.0f / (sqrtf(ss) + 1e-9f);
#pragma unroll
  for (int r = 0; r < 8; ++r) { re[r] *= inv; im[r] *= inv; }
  CS cs;
  __sincosf(0.5f * qff[lane], &cs.s0, &cs.c0);
  cs.c1 = cs.c0; cs.s1 = cs.s0;
  sim_layer<0>(re, im, cs, lane);
  __shared__ float sexp[24];
#define PW_(w) { float xr, xi, zz; pauli<7 - (w)>(re, im, lane, xr, xi, zz); \
                 if (lane == 0) { sexp[(w)] = 2.f * xr; sexp[8 + (w)] = 2.f * xi; sexp[16 + (w)] = zz; } }
  FOR8(PW_)
#undef PW_
  __syncthreads();
  for (int o = lane; o < 128; o += 32) {
    float a = qout_b[o];
#pragma unroll
    for (int k = 0; k < 24; ++k) a = fmaf(sexp[k], qout_w[k * 128 + o], a);
    temporal[b * 128 + o] = a;
  }
}

// ================================================================ K10a: per-(b,c) mean / 1/std over T (ddof=1)
__global__ void __launch_bounds__(256) k_cstats(const float* __restrict__ x,
                                                float* __restrict__ muA, float* __restrict__ rsA) {
  int lane = threadIdx.x & 31;
  int wid = blockIdx.x * 8 + (threadIdx.x >> 5);   // 0..2879
  int b = wid / NC, c = wid % NC;
  float s1 = 0.f, s2 = 0.f;
  for (int t = lane; t < NT; t += 32) {
    float v = x[((size_t)b * NT + t) * NC + c];
    s1 += v; s2 += v * v;
  }
  s1 = wred32(s1); s2 = wred32(s2);
  if (lane == 0) {
    float mu = s1 / (float)NT;
    float var = (s2 - (float)NT * mu * mu) / (float)(NT - 1);
    float sd = fmaxf(sqrtf(fmaxf(var, 0.f)), 1e-8f);
    muA[wid] = mu;
    rsA[wid] = 1.0f / sd;
  }
}

// ================================================================ K10b: block-sum correlation -> triu(12) -> fc head
__global__ void __launch_bounds__(256) k_corr(const float* __restrict__ x,
                                              const float* __restrict__ muA,
                                              const float* __restrict__ rsA,
                                              const float* __restrict__ fc_w,   // (78,64)
                                              const float* __restrict__ fc_b,
                                              float* __restrict__ fcOut) {      // (B,64)
  __shared__ float smu[NC], srs[NC];
  __shared__ float sbuf[256][12];
  __shared__ float stri[78];
  int b = blockIdx.x;
  int tid = threadIdx.x;
  for (int i = tid; i < NC; i += 256) { smu[i] = muA[b * NC + i]; srs[i] = rsA[b * NC + i]; }
  __syncthreads();
  // pair decode for tid < 78 (row-major upper triangle of 12x12)
  int pi = 0, pj = 0;
  {
    int p = tid, rowlen = 12, i = 0;
    while (p >= rowlen && rowlen > 0) { p -= rowlen; rowlen--; i++; }
    pi = i; pj = i + p;
  }
  float triA = 0.f;
  for (int base = 0; base < NT; base += 256) {
    int t = base + tid;
    float s[12];
    if (t < NT) {
      const float* xr = x + ((size_t)b * NT + t) * NC;
#pragma unroll
      for (int i = 0; i < 12; ++i) {
        float a = 0.f;
#pragma unroll
        for (int u = 0; u < 15; ++u) {
          int c = i * 15 + u;
          a += (xr[c] - smu[c]) * srs[c];
        }
        s[i] = a;
      }
    } else {
#pragma unroll
      for (int i = 0; i < 12; ++i) s[i] = 0.f;
    }
#pragma unroll
    for (int i = 0; i < 12; ++i) sbuf[tid][i] = s[i];
    __syncthreads();
    if (tid < 78) {
      float a = 0.f;
      for (int tt = 0; tt < 256; ++tt) a = fmaf(sbuf[tt][pi], sbuf[tt][pj], a);
      triA += a;
    }
    __syncthreads();
  }
  if (tid < 78) stri[tid] = triA * (1.0f / (225.0f * (float)(NT - 1)));
  __syncthreads();
  if (tid < 64) {
    float a = fc_b[tid];
    for (int p = 0; p < 78; ++p) a = fmaf(stri[p], fc_w[p * 64 + tid], a);
    fcOut[b * 64 + tid] = gelu_f(a);
  }
}

// ================================================================ K11: fusion head
__global__ void __launch_bounds__(128) k_head(const float* __restrict__ temporal,  // (B,128)
                                              const float* __restrict__ fcIn,      // (B,64)
                                              const float* __restrict__ f1w,       // (192,128)
                                              const float* __restrict__ f1b,
                                              const float* __restrict__ flng,
                                              const float* __restrict__ flnb,
                                              const float* __restrict__ f2w,       // (128,64)
                                              const float* __restrict__ f2b,
                                              float* __restrict__ out) {           // (B,64)
  __shared__ float fz[192], z[128], red[128];
  int b = blockIdx.x;
  int tid = threadIdx.x;
  fz[tid] = temporal[b * 128 + tid];
  if (tid < 64) fz[128 + tid] = fcIn[b * 64 + tid];
  __syncthreads();
  float acc = f1b[tid];
  for (int k = 0; k < 192; ++k) acc = fmaf(fz[k], f1w[k * 128 + tid], acc);
  // LN stats over 128
  red[tid] = acc; __syncthreads();
  for (int s = 64; s > 0; s >>= 1) { if (tid < s) red[tid] += red[tid + s]; __syncthreads(); }
  float mu = red[0] * (1.f / 128.f);
  __syncthreads();
  red[tid] = acc * acc; __syncthreads();
  for (int s = 64; s > 0; s >>= 1) { if (tid < s) red[tid] += red[tid + s]; __syncthreads(); }
  float var = red[0] * (1.f / 128.f) - mu * mu;
  __syncthreads();
  float rs = rsqrtf(var + 1e-5f);
  z[tid] = gelu_f((acc - mu) * rs * flng[tid] + flnb[tid]);
  __syncthreads();
  if (tid < 64) {
    float o = f2b[tid];
    for (int k = 0; k < 128; ++k) o = fmaf(z[k], f2w[k * 64 + tid], o);
    out[b * 64 + tid] = o;
  }
}

// ================================================================ launch
extern "C" void kernel_launch(void* const* d_in, const int* in_sizes, int n_in,
                              void* d_out, int out_size, void* d_ws, size_t ws_size,
                              hipStream_t stream) {
  const float* x        = (const float*)d_in[0];
  const float* net_w    = (const float*)d_in[1];
  const float* net_b    = (const float*)d_in[2];
  const float* dww[3]   = {(const float*)d_in[3],  (const float*)d_in[9],  (const float*)d_in[15]};
  const float* dwb[3]   = {(const float*)d_in[4],  (const float*)d_in[10], (const float*)d_in[16]};
  const float* pww[3]   = {(const float*)d_in[5],  (const float*)d_in[11], (const float*)d_in[17]};
  const float* pwb[3]   = {(const float*)d_in[6],  (const float*)d_in[12], (const float*)d_in[18]};
  const float* gng[3]   = {(const float*)d_in[7],  (const float*)d_in[13], (const float*)d_in[19]};
  const float* gnb[3]   = {(const float*)d_in[8],  (const float*)d_in[14], (const float*)d_in[20]};
  const float* merge_w  = (const float*)d_in[21];
  const float* merge_b  = (const float*)d_in[22];
  const float* mlng     = (const float*)d_in[23];
  const float* mlnb     = (const float*)d_in[24];
  const float* qproj_w  = (const float*)d_in[25];
  const float* qproj_b  = (const float*)d_in[26];
  const float* poly     = (const float*)d_in[27];
  const float* qff      = (const float*)d_in[28];
  const float* mixre    = (const float*)d_in[29];
  const float* mixim    = (const float*)d_in[30];
  const float* qout_w   = (const float*)d_in[31];
  const float* qout_b   = (const float*)d_in[32];
  const float* fc_w     = (const float*)d_in[33];
  const float* fc_b     = (const float*)d_in[34];
  const float* f1w      = (const float*)d_in[35];
  const float* f1b      = (const float*)d_in[36];
  const float* flng     = (const float*)d_in[37];
  const float* flnb     = (const float*)d_in[38];
  const float* f2w      = (const float*)d_in[39];
  const float* f2b      = (const float*)d_in[40];

  float* ws = (float*)d_ws;
  const size_t o_h   = 0;
  const size_t o_dw  = o_h   + (size_t)BT * 96;
  const size_t o_ypw = o_dw  + (size_t)BT * 96;
  const size_t o_cat = o_ypw + (size_t)BT * 96;
  const size_t o_pwT = o_cat + (size_t)BT * 288;
  const size_t o_gn  = o_pwT + 96 * 96;
  const size_t o_part = o_gn + NB * 8 * 2;
  const size_t o_tmp  = o_part + (size_t)NB * NWV * 512;
  const size_t o_fc   = o_tmp + NB * 128;
  const size_t o_mu   = o_fc + NB * 64;
  const size_t o_rs   = o_mu + NB * NC;
  const size_t o_hm   = o_dw;   // alias: dw/ypw free after merge input (cat) is built
  const size_t o_tp   = o_h;    // alias: h free after conv branches

  const int dils[3] = {1, 4, 16};
  const int gridE = (BT * 96) / 256;   // 2178 exact

  k_stage1<<<gridE, 256, 0, stream>>>(x, net_w, net_b, ws + o_h);
  for (int br = 0; br < 3; ++br) {
    k_dw<<<gridE, 256, 0, stream>>>(ws + o_h, dww[br], dwb[br], dils[br], ws + o_dw);
    k_tr96<<<36, 256, 0, stream>>>(pww[br], ws + o_pwT);
    k_pw<<<BT / 16, 192, 0, stream>>>(ws + o_dw, ws + o_pwT, pwb[br], ws + o_ypw);
    k_gnstats<<<16, 256, 0, stream>>>(ws + o_ypw, ws + o_gn);
    k_gnapply<<<gridE, 256, 0, stream>>>(ws + o_ypw, ws + o_gn, gng[br], gnb[br], br * 96, ws + o_cat);
  }
  k_merge<<<BT / 16, 256, 0, stream>>>(ws + o_cat, merge_w, merge_b, mlng, mlnb, ws + o_hm);
  k_qproj<<<BT / 16, 128, 0, stream>>>(ws + o_hm, qproj_w, qproj_b, ws + o_tp);
  k_quantum<<<NB * GPB, 256, 0, stream>>>(ws + o_tp, poly, mixre, mixim, ws + o_part);
  k_qfinal<<<NB, 32, 0, stream>>>(ws + o_part, qff, qout_w, qout_b, ws + o_tmp);
  k_cstats<<<(NB * NC) / 8, 256, 0, stream>>>(x, ws + o_mu, ws + o_rs);
  k_corr<<<NB, 256, 0, stream>>>(x, ws + o_mu, ws + o_rs, fc_w, fc_b, ws + o_fc);
  k_head<<<NB, 128, 0, stream>>>(ws + o_tmp, ws + o_fc, f1w, f1b, flng, flnb, f2w, f2b, (float*)d_out);
}